// DistanceDecayAttention_61899068670212
// MI455X (gfx1250) — compile-verified
//
#include <hip/hip_runtime.h>
#include <hip/hip_bf16.h>
#include <math.h>

// Problem constants from the reference: B=256, N=1024, E=16384, DECAY=0.2
#define NN        1024
#define EE        16384
#define NT        1024            // 32 wave32s per block
#define INF_BITS  0x7F800000

// Low 32 bits of a generic pointer to __shared__ = workgroup-relative LDS
// byte offset (flat-LDS aperture encoding; HW adds LDS_BASE itself).
__device__ __forceinline__ unsigned lds_off(const void* p) {
    return (unsigned)(unsigned long long)p;
}

// CDNA5 async copy: 16 bytes per lane, memory -> LDS, bypassing VGPRs.
// Tracked by ASYNCcnt; completed with s_wait_asynccnt.
__device__ __forceinline__ void async_ld_b128(unsigned lofs, const void* gaddr) {
    asm volatile("global_load_async_to_lds_b128 %0, %1, off"
                 :: "v"(lofs), "v"(gaddr) : "memory");
}

__device__ __forceinline__ void wait_async_zero() {
    asm volatile("s_wait_asynccnt 0x0" ::: "memory");
}

__global__ __launch_bounds__(NT, 1)
void sssp_decay_kernel(const int*   __restrict__ edge_index,  // (B,2,E)
                       const float* __restrict__ edge_attr,   // (B,E)
                       const int*   __restrict__ p_node_id,   // (B,)
                       const float* __restrict__ logits,      // (B,N)
                       float*       __restrict__ out)         // (B,N)
{
    // Whole graph lives in LDS: 64KB + 64KB + 64KB edges + 4KB dist = 196KB
    // (fits the 320KB/WGP CDNA5 LDS; one workgroup per WGP).
    __shared__ __align__(16) int   s_src[EE];
    __shared__ __align__(16) int   s_dst[EE];
    __shared__ __align__(16) float s_w[EE];
    __shared__ __align__(16) int   s_dist[NN];  // float bit patterns; all >= 0
    __shared__ int   s_changed;

    const int b   = blockIdx.x;
    const int tid = threadIdx.x;

    const int*   g_src = edge_index + (size_t)b * 2 * EE;   // edge_index[b][0][:]
    const int*   g_dst = g_src + EE;                        // edge_index[b][1][:]
    const float* g_w   = edge_attr + (size_t)b * EE;

    // ---- Stage the full edge list into LDS with async B128 copies ----
    // EE/4 = 4096 16B chunks per array; 1024 lanes -> 4 passes per array.
    for (int base = tid * 4; base < EE; base += NT * 4) {
        async_ld_b128(lds_off(&s_src[base]), g_src + base);
        async_ld_b128(lds_off(&s_dst[base]), g_dst + base);
        async_ld_b128(lds_off(&s_w  [base]), g_w   + base);
    }

    // Initialize distances to +inf while the async copies are in flight
    // (disjoint LDS region).
    for (int i = tid; i < NN; i += NT) s_dist[i] = INF_BITS;

    wait_async_zero();     // this wave's async copies landed in LDS
    __syncthreads();       // all waves' copies visible block-wide

    // Reference semantics: edges with latency == 1.0 are skipped -> w = +inf.
    const float inf = __int_as_float(INF_BITS);
    for (int e = tid; e < EE; e += NT) {
        if (s_w[e] == 1.0f) s_w[e] = inf;
    }
    if (tid == 0) s_dist[p_node_id[b]] = 0;   // float 0.0f bits == 0
    __syncthreads();

    // ---- Bellman-Ford scatter-min, entirely in LDS ----
    // Non-negative floats (incl. +inf) order identically as signed ints,
    // so ds_min_i32 on the bit pattern is an exact float atomic-min.
    // Edge stream is read with ds_load_b128 (4 edges per thread per trip).
    for (int it = 0; it < NN; ++it) {
        if (tid == 0) s_changed = 0;
        __syncthreads();

        bool ch = false;
        #pragma unroll
        for (int e0 = tid * 4; e0 < EE; e0 += NT * 4) {
            const int4   s4 = *(const int4*  )(s_src + e0);
            const int4   d4 = *(const int4*  )(s_dst + e0);
            const float4 w4 = *(const float4*)(s_w   + e0);
            const int   ss[4] = { s4.x, s4.y, s4.z, s4.w };
            const int   dd_[4]= { d4.x, d4.y, d4.z, d4.w };
            const float ww[4] = { w4.x, w4.y, w4.z, w4.w };
            #pragma unroll
            for (int k = 0; k < 4; ++k) {
                const int   s  = ss[k];
                const int   d  = dd_[k];
                const float wf = ww[k];
                const float ds = __int_as_float(s_dist[s]);
                const float dd = __int_as_float(s_dist[d]);
                const float c1 = ds + wf;           // relax s -> d
                if (c1 < dd) { atomicMin(&s_dist[d], __float_as_int(c1)); ch = true; }
                const float c2 = dd + wf;           // relax d -> s (undirected)
                if (c2 < ds) { atomicMin(&s_dist[s], __float_as_int(c2)); ch = true; }
            }
        }
        if (ch) s_changed = 1;
        __syncthreads();
        if (!s_changed) break;   // uniform across block
        __syncthreads();         // protect s_changed reset next iteration
    }

    // ---- decay = exp(-0.2*dist), inf -> 1e-8; out = logits * decay ----
    const float* lg = logits + (size_t)b * NN;
    float*       op = out    + (size_t)b * NN;
    for (int i = tid; i < NN; i += NT) {
        const float df  = __int_as_float(s_dist[i]);
        const float dec = (df < inf) ? expf(-0.2f * df) : 1e-8f;
        op[i] = lg[i] * dec;
    }
}

extern "C" void kernel_launch(void* const* d_in, const int* in_sizes, int n_in,
                              void* d_out, int out_size, void* d_ws, size_t ws_size,
                              hipStream_t stream) {
    const int*   edge_index = (const int*)  d_in[0];   // (B,2,E) int32
    const float* edge_attr  = (const float*)d_in[1];   // (B,E)   f32
    const int*   p_node_id  = (const int*)  d_in[2];   // (B,)    int32
    const float* logits     = (const float*)d_in[3];   // (B,N)   f32
    float*       out        = (float*)d_out;           // (B,N)   f32

    const int B = in_sizes[2];                         // 256
    sssp_decay_kernel<<<B, NT, 0, stream>>>(edge_index, edge_attr,
                                            p_node_id, logits, out);
}